// SGC_69114613730233
// MI455X (gfx1250) — compile-verified
//
#include <hip/hip_runtime.h>

#define N_NODES 100000
#define D_IN    128
#define D_OUT   64

typedef __attribute__((ext_vector_type(2))) float v2f;
typedef __attribute__((ext_vector_type(4))) float v4f;
typedef __attribute__((ext_vector_type(8))) float v8f;

// ---------------- workspace init ----------------
__global__ void sgc_zero_kernel(float* ws, int total) {
    int i = blockIdx.x * blockDim.x + threadIdx.x;
    if (i < total) ws[i] = 0.0f;
}

// ---------------- degree count: deg[row[e]] += 1 ----------------
__global__ void sgc_degree_kernel(const int* __restrict__ row, float* __restrict__ deg, int E) {
    int e = blockIdx.x * blockDim.x + threadIdx.x;
    if (e < E) atomicAdd(&deg[row[e]], 1.0f);
}

// ---------------- dinv[i] = (deg[i] + 1)^-1/2  (self loop adds 1) ----------------
__global__ void sgc_dinv_kernel(float* __restrict__ deg) {
    int i = blockIdx.x * blockDim.x + threadIdx.x;
    if (i < N_NODES) deg[i] = rsqrtf(deg[i] + 1.0f);
}

// ---------------- edge scatter: support[r] += (dinv[r]*dinv[c])^k * x[c] ----------------
// 32 threads per edge, each handles 4 contiguous features (float4 gather, 4 atomics).
// x (51 MB) is L2-resident on MI455X (192 MB L2), so gathers + atomics stay in L2.
__global__ void sgc_scatter_kernel(const int* __restrict__ row, const int* __restrict__ col,
                                   const float* __restrict__ x, const float* __restrict__ dinv,
                                   float* __restrict__ support, const int* __restrict__ kptr,
                                   int E) {
    long long tid = (long long)blockIdx.x * blockDim.x + threadIdx.x;
    int lane = (int)(tid & 31);
    long long e = tid >> 5;
    if (e >= E) return;
    int r = row[e];
    int c = col[e];
    float kf = (float)(*kptr);
    float w = __powf(dinv[r] * dinv[c], kf);
    const v4f xv = *(const v4f*)(x + (long long)c * D_IN + lane * 4);
    float* s = support + (long long)r * D_IN + lane * 4;
    atomicAdd(s + 0, w * xv.x);
    atomicAdd(s + 1, w * xv.y);
    atomicAdd(s + 2, w * xv.z);
    atomicAdd(s + 3, w * xv.w);
}

// ---------------- self loop: support[i] += (dinv[i]^2)^k * x[i] ----------------
__global__ void sgc_selfloop_kernel(const float* __restrict__ x, const float* __restrict__ dinv,
                                    float* __restrict__ support, const int* __restrict__ kptr) {
    int i = blockIdx.x * blockDim.x + threadIdx.x;   // over N*D_IN = 12.8M
    if (i >= N_NODES * D_IN) return;
    int node = i >> 7;
    float dv = dinv[node];
    float kf = (float)(*kptr);
    float w = __powf(dv * dv, kf);
    support[i] += w * x[i];
}

// ---------------- dense projection via V_WMMA_F32_16X16X4_F32 ----------------
// Block = 128 threads = 4 waves. Block handles 16 node-rows; each wave owns one
// 16-wide slice of D_OUT. K=128 consumed in 32 steps of 4 (f32 WMMA).
// A layout (16x4 f32): lanes 0-15 -> M=lane, VGPR0=K0 VGPR1=K1; lanes 16-31 -> M=lane-16, K2/K3.
// B layout (4x16 f32): symmetric with N in place of M.
// C/D layout (16x16 f32): VGPR r: lanes 0-15 -> M=r, N=lane; lanes 16-31 -> M=8+r, N=lane-16.
__global__ __launch_bounds__(128) void sgc_gemm_wmma_kernel(const float* __restrict__ support,
                                                            const float* __restrict__ W,
                                                            const float* __restrict__ bias,
                                                            float* __restrict__ out) {
    int wave = threadIdx.x >> 5;          // 0..3 -> D_OUT tile
    int lane = threadIdx.x & 31;
    int row0 = blockIdx.x * 16;           // 6250 blocks * 16 = 100000 exactly
    int n0   = wave * 16;
    int m    = lane & 15;                 // M index for A, N index for B
    int kk   = (lane >> 4) * 2;           // K sub-offset within the 4-wide step

    const float* arow = support + (long long)(row0 + m) * D_IN + kk; // A[M][K]
    const float* brow = W + (long long)(n0 + m) * D_IN + kk;         // B[K][N] = W[n][k]

    v8f acc = {};
    #pragma unroll
    for (int k0 = 0; k0 < D_IN; k0 += 4) {
        v2f a = *(const v2f*)(arow + k0);
        v2f b = *(const v2f*)(brow + k0);
        acc = __builtin_amdgcn_wmma_f32_16x16x4_f32(
            /*neg_a=*/false, a, /*neg_b=*/false, b,
            /*c_mod=*/(short)0, acc, /*reuse_a=*/false, /*reuse_b=*/false);
    }

    float bv = bias[n0 + m];
    int mbase = row0 + ((lane >> 4) * 8);
    #pragma unroll
    for (int r = 0; r < 8; ++r) {
        out[(long long)(mbase + r) * D_OUT + n0 + m] = acc[r] + bv;
    }
}

extern "C" void kernel_launch(void* const* d_in, const int* in_sizes, int n_in,
                              void* d_out, int out_size, void* d_ws, size_t ws_size,
                              hipStream_t stream) {
    const float* x    = (const float*)d_in[0];
    const int*   edge = (const int*)d_in[1];   // [2, E] flat
    const float* W    = (const float*)d_in[2]; // [D_OUT, D_IN]
    const float* bias = (const float*)d_in[3];
    const int*   kptr = (const int*)d_in[4];
    int E = in_sizes[1] / 2;

    float* deg     = (float*)d_ws;           // [N] -> becomes dinv in place
    float* support = deg + N_NODES;          // [N, D_IN]

    const int* row = edge;
    const int* col = edge + E;

    int totalZero = N_NODES * (1 + D_IN);
    sgc_zero_kernel<<<(totalZero + 255) / 256, 256, 0, stream>>>(deg, totalZero);

    sgc_degree_kernel<<<(E + 255) / 256, 256, 0, stream>>>(row, deg, E);

    sgc_dinv_kernel<<<(N_NODES + 255) / 256, 256, 0, stream>>>(deg);

    long long scatterThreads = (long long)E * 32;
    sgc_scatter_kernel<<<(unsigned)((scatterThreads + 255) / 256), 256, 0, stream>>>(
        row, col, x, deg, support, kptr, E);

    sgc_selfloop_kernel<<<(N_NODES * D_IN + 255) / 256, 256, 0, stream>>>(x, deg, support, kptr);

    sgc_gemm_wmma_kernel<<<N_NODES / 16, 128, 0, stream>>>(support, W, bias, (float*)d_out);
}